// BasicTransformerBlock_18657337934637
// MI455X (gfx1250) — compile-verified
//
#include <hip/hip_runtime.h>
#include <hip/hip_bf16.h>

typedef float v2f __attribute__((ext_vector_type(2)));
typedef float v8f __attribute__((ext_vector_type(8)));

#define SEQ   4096
#define DIM   2304
#define HEADS 24
#define HD    96
#define GLEN  1024                      // keys per sparse batch (SEQ/4)
#define ATT_SCALE 0.10206207261596577f  // 1/sqrt(96)

// ---------------------------------------------------------------------------
// f32 WMMA 16x16x4:  D = A(16x4) * B(4x16) + C(16x16), full fp32 precision.
// ---------------------------------------------------------------------------
__device__ __forceinline__ v8f wmma_f32(v2f a, v2f b, v8f c) {
  return __builtin_amdgcn_wmma_f32_16x16x4_f32(
      /*neg_a=*/false, a, /*neg_b=*/false, b,
      /*c_mod=*/(short)0, c, /*reuse_a=*/false, /*reuse_b=*/false);
}

// ---------------------------------------------------------------------------
// GEMM + bias:  C[M,N] = A[M,K] @ B[K,N] + bias[N]    (row-major everywhere)
// 256 threads = 8 waves (wave grid 2m x 4n); block tile 128x128;
// wave tile 64x32 (4x2 WMMA accumulators = 64 VGPRs).
// Software-pipelined: next global tile staged in registers during compute.
// ---------------------------------------------------------------------------
__global__ __launch_bounds__(256)
void gemm_bias_kernel(const float* __restrict__ A, const float* __restrict__ B,
                      const float* __restrict__ bias, float* __restrict__ C,
                      int M, int N, int K) {
  __shared__ float As[128 * 17];   // 128 rows x 16 k, pad 17 (gcd(17,64)=1)
  __shared__ float Bs[16 * 132];   // 16 k x 128 cols, pad 132

  const int tid  = threadIdx.x;
  const int wave = tid >> 5, lane = tid & 31;
  const int half = lane >> 4, l16 = lane & 15;
  const int wm = wave >> 2, wn = wave & 3;       // 2 x 4 wave grid
  const int m0 = blockIdx.y * 128, n0 = blockIdx.x * 128;

  v8f acc[4][2] = {};

  // Loaders: 256 threads, each stages 2 float4 of A and 2 float4 of B.
  const int ar = tid >> 1, ac = (tid & 1) * 8;    // A: 128 rows x 16 k
  const int br = tid >> 4, bc = (tid & 15) * 8;   // B: 16 k  x 128 cols

  const float* aptr = A + (size_t)(m0 + ar) * K + ac;
  const float* bptr = B + (size_t)br * N + n0 + bc;

  float4 a0v = *(const float4*)(aptr + 0);
  float4 a1v = *(const float4*)(aptr + 4);
  float4 b0v = *(const float4*)(bptr + 0);
  float4 b1v = *(const float4*)(bptr + 4);

  for (int kb = 0; kb < K; kb += 16) {
    As[ar * 17 + ac + 0] = a0v.x; As[ar * 17 + ac + 1] = a0v.y;
    As[ar * 17 + ac + 2] = a0v.z; As[ar * 17 + ac + 3] = a0v.w;
    As[ar * 17 + ac + 4] = a1v.x; As[ar * 17 + ac + 5] = a1v.y;
    As[ar * 17 + ac + 6] = a1v.z; As[ar * 17 + ac + 7] = a1v.w;
    Bs[br * 132 + bc + 0] = b0v.x; Bs[br * 132 + bc + 1] = b0v.y;
    Bs[br * 132 + bc + 2] = b0v.z; Bs[br * 132 + bc + 3] = b0v.w;
    Bs[br * 132 + bc + 4] = b1v.x; Bs[br * 132 + bc + 5] = b1v.y;
    Bs[br * 132 + bc + 6] = b1v.z; Bs[br * 132 + bc + 7] = b1v.w;
    __syncthreads();

    if (kb + 16 < K) {   // stage next tile; waited on at next LDS store
      const float* an = aptr + kb + 16;
      const float* bn = bptr + (size_t)(kb + 16) * N;
      __builtin_prefetch(an + 64, 0, 3);
      a0v = *(const float4*)(an + 0);
      a1v = *(const float4*)(an + 4);
      b0v = *(const float4*)(bn + 0);
      b1v = *(const float4*)(bn + 4);
    }

#pragma unroll
    for (int kk = 0; kk < 16; kk += 4) {
      v2f af[4], bf[2];
#pragma unroll
      for (int s = 0; s < 4; ++s) {
        const float* ap = &As[(wm * 64 + s * 16 + l16) * 17 + kk + 2 * half];
        af[s][0] = ap[0];  af[s][1] = ap[1];
      }
#pragma unroll
      for (int t = 0; t < 2; ++t) {
        const float* bp = &Bs[(kk + 2 * half) * 132 + wn * 32 + t * 16 + l16];
        bf[t][0] = bp[0];  bf[t][1] = bp[132];
      }
#pragma unroll
      for (int s = 0; s < 4; ++s)
#pragma unroll
        for (int t = 0; t < 2; ++t)
          acc[s][t] = wmma_f32(af[s], bf[t], acc[s][t]);
    }
    __syncthreads();
  }

#pragma unroll
  for (int s = 0; s < 4; ++s)
#pragma unroll
    for (int t = 0; t < 2; ++t) {
      const int col = n0 + wn * 32 + t * 16 + l16;
      const float bb = bias[col];
#pragma unroll
      for (int e = 0; e < 8; ++e) {
        const int row = m0 + wm * 64 + s * 16 + 8 * half + e;
        C[(size_t)row * N + col] = acc[s][t][e] + bb;
      }
    }
}

// ---------------------------------------------------------------------------
// 3D RoPE, in place on [SEQ][DIM].  One thread per (token, head, part, pair).
// ---------------------------------------------------------------------------
__global__ void rope_kernel(float* __restrict__ x, int total) {
  int idx = blockIdx.x * blockDim.x + threadIdx.x;
  if (idx >= total) return;
  const int i    = idx & 15;        // freq index 0..15
  int rest       = idx >> 4;
  const int part = rest % 3;  rest /= 3;
  const int head = rest % HEADS;
  const int tok  = rest / HEADS;
  const int tpos = tok >> 9;              // frame  (token / (16*32))
  const int ypos = (tok >> 5) & 15;       // height
  const int xpos = tok & 31;              // width
  const int pos  = (part == 0) ? tpos : ((part == 1) ? ypos : xpos);
  const float ang = (float)pos * powf(10000.0f, -(float)i / 16.0f);
  float s, c;
  sincosf(ang, &s, &c);
  const size_t base = (size_t)tok * DIM + head * HD + part * 32 + i;
  const float x1 = x[base], x2 = x[base + 16];
  x[base]      = x1 * c - x2 * s;
  x[base + 16] = x2 * c + x1 * s;
}

// ---------------------------------------------------------------------------
// Sparse attention.  grid = (16 q-tiles, 24 heads, 4 sparse batches).
// Token of sparse position j (batch nidx) = 4*j + nidx.
// LDS (dynamic, 312 KB of the 320 KB WGP LDS):
//   S  : 64 x 1025  fp32 score panel (full 1024-key row -> 2-pass softmax,
//                   stride 1025 == 1 mod 64 -> conflict-free column walks)
//   Qs : 64 x 97    query tile
//   KVs: 64 x 97    streamed K (phase A) / V (phase C) tile
// ---------------------------------------------------------------------------
__device__ __forceinline__
void load_tile96(float* __restrict__ dst, const float* __restrict__ src,
                 int rowbase, int nidx, int head) {
  const int t = threadIdx.x;          // 512 threads: 8 per row, 12 floats each
  const int r = t >> 3;
  const int c = (t & 7) * 12;
  const int token = 4 * (rowbase + r) + nidx;
  const float* s = src + (size_t)token * DIM + head * HD + c;
  float* d = dst + r * 97 + c;
#pragma unroll
  for (int i = 0; i < 3; ++i) {
    float4 tmp = ((const float4*)s)[i];
    d[i * 4 + 0] = tmp.x; d[i * 4 + 1] = tmp.y;
    d[i * 4 + 2] = tmp.z; d[i * 4 + 3] = tmp.w;
  }
}

__global__ __launch_bounds__(512)
void attention_kernel(const float* __restrict__ q, const float* __restrict__ k,
                      const float* __restrict__ v, float* __restrict__ o) {
  extern __shared__ float smem[];
  float* S   = smem;               // 64*1025
  float* Qs  = S + 64 * 1025;      // 64*97
  float* KVs = Qs + 64 * 97;       // 64*97

  const int qt = blockIdx.x, head = blockIdx.y, nidx = blockIdx.z;
  const int tid  = threadIdx.x;
  const int wave = tid >> 5, lane = tid & 31;
  const int half = lane >> 4, l16 = lane & 15;

  load_tile96(Qs, q, qt * 64, nidx, head);

  // ---- Phase A: S = scale * Q @ K^T  (one 16x16 tile per wave per K-block)
  const int mi = wave & 3, nj = wave >> 2;
  for (int kb = 0; kb < 16; ++kb) {
    load_tile96(KVs, k, kb * 64, nidx, head);
    __syncthreads();
    v8f acc = {};
    const float* qrow = &Qs[(mi * 16 + l16) * 97];
    const float* krow = &KVs[(nj * 16 + l16) * 97];
#pragma unroll
    for (int kk = 0; kk < HD; kk += 4) {
      v2f a, b;
      a[0] = qrow[kk + 2 * half]; a[1] = qrow[kk + 2 * half + 1];
      b[0] = krow[kk + 2 * half]; b[1] = krow[kk + 2 * half + 1];
      acc = wmma_f32(a, b, acc);
    }
    float* srow = &S[(size_t)(mi * 16 + 8 * half) * 1025 + kb * 64 + nj * 16 + l16];
#pragma unroll
    for (int j = 0; j < 8; ++j) srow[(size_t)j * 1025] = acc[j] * ATT_SCALE;
    __syncthreads();
  }

  // ---- Phase B: softmax over each 1024-wide row (one wave per row, wave32)
  for (int rr = 0; rr < 4; ++rr) {
    const int row = wave + rr * 16;
    float* srow = &S[(size_t)row * 1025];
    float vals[32];
    float m = -3.402823466e+38f;
#pragma unroll
    for (int i = 0; i < 32; ++i) {
      vals[i] = srow[lane + i * 32];
      m = fmaxf(m, vals[i]);
    }
#pragma unroll
    for (int off = 16; off > 0; off >>= 1) m = fmaxf(m, __shfl_xor(m, off, 32));
    float sum = 0.0f;
#pragma unroll
    for (int i = 0; i < 32; ++i) { vals[i] = expf(vals[i] - m); sum += vals[i]; }
#pragma unroll
    for (int off = 16; off > 0; off >>= 1) sum += __shfl_xor(sum, off, 32);
    const float inv = 1.0f / sum;
#pragma unroll
    for (int i = 0; i < 32; ++i) srow[lane + i * 32] = vals[i] * inv;
  }
  __syncthreads();

  // ---- Phase C: O = P @ V.  24 output tiles; wave w owns (mi, dj0=w>>2) and,
  // for w<8, (mi, dj1=dj0+4).  Both tiles share the SAME row block mi, so one
  // A-fragment (P rows) feeds two WMMAs.
  const int dj0 = wave >> 2;
  const int dj1 = dj0 + 4;
  v8f o0 = {}, o1 = {};
  for (int kb = 0; kb < 16; ++kb) {
    load_tile96(KVs, v, kb * 64, nidx, head);
    __syncthreads();
#pragma unroll
    for (int kk = 0; kk < 64; kk += 4) {
      const int kcol = kb * 64 + kk + 2 * half;
      const float* prow = &S[(size_t)(mi * 16 + l16) * 1025 + kcol];
      v2f a, b0;
      a[0] = prow[0]; a[1] = prow[1];
      b0[0] = KVs[(kk + 2 * half) * 97 + dj0 * 16 + l16];
      b0[1] = KVs[(kk + 2 * half + 1) * 97 + dj0 * 16 + l16];
      o0 = wmma_f32(a, b0, o0);
      if (wave < 8) {
        v2f b1;
        b1[0] = KVs[(kk + 2 * half) * 97 + dj1 * 16 + l16];
        b1[1] = KVs[(kk + 2 * half + 1) * 97 + dj1 * 16 + l16];
        o1 = wmma_f32(a, b1, o1);
      }
    }
    __syncthreads();
  }

  // Write back with reverse-sparse indexing fused (token = 4*pos + nidx).
#pragma unroll
  for (int j = 0; j < 8; ++j) {
    const int m = mi * 16 + 8 * half + j;
    const int token = 4 * (qt * 64 + m) + nidx;
    o[(size_t)token * DIM + head * HD + dj0 * 16 + l16] = o0[j];
  }
  if (wave < 8) {
#pragma unroll
    for (int j = 0; j < 8; ++j) {
      const int m = mi * 16 + 8 * half + j;
      const int token = 4 * (qt * 64 + m) + nidx;
      o[(size_t)token * DIM + head * HD + dj1 * 16 + l16] = o1[j];
    }
  }
}

// ---------------------------------------------------------------------------
extern "C" void kernel_launch(void* const* d_in, const int* in_sizes, int n_in,
                              void* d_out, int out_size, void* d_ws, size_t ws_size,
                              hipStream_t stream) {
  const float* h  = (const float*)d_in[0];
  const float* wq = (const float*)d_in[1];
  const float* bq = (const float*)d_in[2];
  const float* wk = (const float*)d_in[3];
  const float* bk = (const float*)d_in[4];
  const float* wv = (const float*)d_in[5];
  const float* bv = (const float*)d_in[6];
  const float* wo = (const float*)d_in[7];
  const float* bo = (const float*)d_in[8];
  // d_in[9..11] = frame/height/width (compile-time constants here: 8/16/32)

  const size_t NE = (size_t)SEQ * DIM;
  float* qb = (float*)d_ws;
  float* kb = qb + NE;
  float* vb = kb + NE;
  float* ab = vb + NE;

  dim3 ggrid(DIM / 128, SEQ / 128);   // (18, 32)
  gemm_bias_kernel<<<ggrid, 256, 0, stream>>>(h, wq, bq, qb, SEQ, DIM, DIM);
  gemm_bias_kernel<<<ggrid, 256, 0, stream>>>(h, wk, bk, kb, SEQ, DIM, DIM);
  gemm_bias_kernel<<<ggrid, 256, 0, stream>>>(h, wv, bv, vb, SEQ, DIM, DIM);

  const int rope_total = SEQ * HEADS * 3 * 16;
  rope_kernel<<<(rope_total + 255) / 256, 256, 0, stream>>>(qb, rope_total);
  rope_kernel<<<(rope_total + 255) / 256, 256, 0, stream>>>(kb, rope_total);

  const size_t smem_bytes = (size_t)(64 * 1025 + 2 * 64 * 97) * sizeof(float); // 312064
  hipFuncSetAttribute(reinterpret_cast<const void*>(attention_kernel),
                      hipFuncAttributeMaxDynamicSharedMemorySize, (int)smem_bytes);
  attention_kernel<<<dim3(GLEN / 64, HEADS, 4), 512, smem_bytes, stream>>>(qb, kb, vb, ab);

  gemm_bias_kernel<<<ggrid, 256, 0, stream>>>(ab, wo, bo, (float*)d_out, SEQ, DIM, DIM);
}